// TemporalAttn_86887188398861
// MI455X (gfx1250) — compile-verified
//
#include <hip/hip_runtime.h>
#include <hip/hip_bf16.h>
#include <math.h>

// ---------------------------------------------------------------------------
// TemporalAttn for MI455X (gfx1250): all matmuls on V_WMMA_F32_16X16X4_F32.
// Round 3: K-major LDS panels so every B-fragment is one contiguous
// ds_load_b64 straight into an even-aligned VGPR pair (no v_mov assembly).
//   - wbufT stride 36: banks (36c+kk) mod 64 = 16-lane permutation,
//     lane halves disjoint via kk mod 4 -> conflict-free.
//   - vbufT stride 68: banks (4n+kk) mod 64, same property.
//   K1: per-joint QKV projections -> ws Q/K/V as [b][j][h][T][hd]
//   K2: attention per (head, row-half); K/V chunks + P-transpose via LDS
//   K3: output projection y = O*Wo^T + bo (Wo is already K-major: pure copy)
// Workspace: 4 * 25,165,824 floats = 402.7 MB.
// ---------------------------------------------------------------------------

typedef __attribute__((ext_vector_type(2))) float v2f;
typedef __attribute__((ext_vector_type(8))) float v8f;

__device__ __forceinline__ v8f wmma_f32(v2f a, v2f b, v8f c) {
    return __builtin_amdgcn_wmma_f32_16x16x4_f32(false, a, false, b,
                                                 (short)0, c, false, false);
}

__device__ __forceinline__ v8f zero8() {
    v8f z = {0.f, 0.f, 0.f, 0.f, 0.f, 0.f, 0.f, 0.f};
    return z;
}

#define KST 68   // row stride (floats) for K panel, rows of 64 + pad
#define TST 36   // row stride (floats) for K-major 32-deep W panels
#define VST 68   // row stride (floats) for K-major 64-deep V panel
#define PST 18   // row stride for 16x16 P transpose tiles

// ---------------------------------------------------------------------------
// Kernel 1: q[b,t,j,e] = sum_d x[b,t,j,d] * W[j,d,e]
// grid = (8, 16, 24), block = 256 (8 waves). Wave tile 16M x 64N, K=512.
// W panel staged K-major: wbufT[col][kk], fragments are single b64 loads.
// ---------------------------------------------------------------------------
__global__ void __launch_bounds__(256)
qkv_proj_kernel(const float* __restrict__ x, const float* __restrict__ W,
                float* __restrict__ out) {
    __shared__ float wbufT[64 * TST];   // 9.2 KB, [col][kk]

    const int tid  = threadIdx.x;
    const int lane = tid & 31;
    const int wave = tid >> 5;
    const int kh   = lane >> 4;
    const int l16  = lane & 15;

    const int j  = blockIdx.z;
    const int n0 = blockIdx.x * 64;
    const int m0 = blockIdx.y * 128 + wave * 16;

    const float* Wj   = W + (size_t)j * 512 * 512;
    const float* arow = x + ((size_t)(m0 + l16) * 24 + j) * 512;

    v8f acc[4];
#pragma unroll
    for (int on = 0; on < 4; ++on) acc[on] = zero8();

    const int crow = tid >> 3;          // 0..31 : k row within panel
    const int ccol = (tid & 7) * 8;     // 0..56 : col group of 8

    for (int kc = 0; kc < 512; kc += 32) {
        // transpose-stage: wbufT[col][kk] = W[kc+kk][n0+col]
        {
            const float* g = Wj + (size_t)(kc + crow) * 512 + n0 + ccol;
            float4 t0 = ((const float4*)g)[0];
            float4 t1 = ((const float4*)g)[1];
            wbufT[(ccol + 0) * TST + crow] = t0.x;
            wbufT[(ccol + 1) * TST + crow] = t0.y;
            wbufT[(ccol + 2) * TST + crow] = t0.z;
            wbufT[(ccol + 3) * TST + crow] = t0.w;
            wbufT[(ccol + 4) * TST + crow] = t1.x;
            wbufT[(ccol + 5) * TST + crow] = t1.y;
            wbufT[(ccol + 6) * TST + crow] = t1.z;
            wbufT[(ccol + 7) * TST + crow] = t1.w;
        }
        __syncthreads();
#pragma unroll
        for (int kr = 0; kr < 32; kr += 4) {
            v2f a = *(const v2f*)(arow + kc + kr + 2 * kh);
#pragma unroll
            for (int on = 0; on < 4; ++on) {
                v2f b = *(const v2f*)(wbufT + (on * 16 + l16) * TST + kr + 2 * kh);
                acc[on] = wmma_f32(a, b, acc[on]);
            }
        }
        __syncthreads();
    }

    // store into [b][j][h][t][hd]
#pragma unroll
    for (int on = 0; on < 4; ++on) {
        const int col = n0 + on * 16 + l16;
        const int h = col >> 6, eh = col & 63;
#pragma unroll
        for (int v = 0; v < 8; ++v) {
            const int rr = m0 + kh * 8 + v;
            const int b = rr >> 8, t = rr & 255;
            out[((((size_t)b * 24 + j) * 8 + h) * 256 + t) * 64 + eh] = acc[on][v];
        }
    }
}

// ---------------------------------------------------------------------------
// Kernel 2: attention. grid = 3072 (head * row-half), block = 256 (8 waves).
// kbuf: K rows row-major (fragments already contiguous b64).
// vbufT: V staged K-major so PV B-fragments are contiguous b64.
// ---------------------------------------------------------------------------
__global__ void __launch_bounds__(256)
attn_kernel(const float* __restrict__ Q, const float* __restrict__ K,
            const float* __restrict__ V, float* __restrict__ O) {
    __shared__ float kbuf[64 * KST];        // 17.0 KB  [key row][hd]
    __shared__ float vbufT[64 * VST];       // 17.0 KB  [col n][key row kk]
    __shared__ float pbuf[8][16 * PST];     //  9.0 KB

    const int tid  = threadIdx.x;
    const int lane = tid & 31;
    const int wave = tid >> 5;
    const int kh   = lane >> 4;
    const int l16  = lane & 15;

    const int hid   = blockIdx.x >> 1;      // (b*24 + j)*8 + h
    const int mhalf = blockIdx.x & 1;
    const int b = hid / 192;
    const int rem = hid % 192;
    const int j = rem >> 3, h = rem & 7;

    const float* q = Q + (size_t)hid * 256 * 64;
    const float* k = K + (size_t)hid * 256 * 64;
    const float* v = V + (size_t)hid * 256 * 64;
    float* pw = &pbuf[wave][0];

    const int m0 = mhalf * 128 + wave * 16;

    const int crow = tid >> 2;              // 0..63
    const int ccol = (tid & 3) * 16;        // 0,16,32,48

    // preload Q fragments, folding in the 1/sqrt(hd) scale
    v2f qf[16];
#pragma unroll
    for (int ks = 0; ks < 16; ++ks)
        qf[ks] = *(const v2f*)(q + (size_t)(m0 + l16) * 64 + ks * 4 + 2 * kh) * 0.125f;

    // ---- S = (Q*0.125) K^T ----
    v8f s[16];
#pragma unroll
    for (int nt = 0; nt < 16; ++nt) s[nt] = zero8();

    for (int kc = 0; kc < 4; ++kc) {        // key rows kc*64 .. +63
        {
            const float* g = k + (size_t)kc * 64 * 64 + (size_t)crow * 64 + ccol;
            float* d = kbuf + crow * KST + ccol;
            float4 t0 = ((const float4*)g)[0], t1 = ((const float4*)g)[1];
            float4 t2 = ((const float4*)g)[2], t3 = ((const float4*)g)[3];
            ((float4*)d)[0] = t0; ((float4*)d)[1] = t1;
            ((float4*)d)[2] = t2; ((float4*)d)[3] = t3;
        }
        __syncthreads();
#pragma unroll
        for (int ntl = 0; ntl < 4; ++ntl) {
            const int nt = kc * 4 + ntl;
            v8f acc = s[nt];
#pragma unroll
            for (int ks = 0; ks < 16; ++ks) {
                v2f bf = *(const v2f*)(kbuf + (ntl * 16 + l16) * KST + ks * 4 + 2 * kh);
                acc = wmma_f32(qf[ks], bf, acc);
            }
            s[nt] = acc;
        }
        __syncthreads();
    }

    // ---- row softmax (row lives in one 16-lane group) ----
#pragma unroll
    for (int r = 0; r < 8; ++r) {
        float m = s[0][r];
#pragma unroll
        for (int nt = 1; nt < 16; ++nt) m = fmaxf(m, s[nt][r]);
#pragma unroll
        for (int off = 8; off >= 1; off >>= 1)
            m = fmaxf(m, __shfl_xor(m, off, 32));
        float sum = 0.f;
#pragma unroll
        for (int nt = 0; nt < 16; ++nt) {
            float e = __expf(s[nt][r] - m);
            s[nt][r] = e;
            sum += e;
        }
#pragma unroll
        for (int off = 8; off >= 1; off >>= 1)
            sum += __shfl_xor(sum, off, 32);
        const float inv = 1.f / sum;
#pragma unroll
        for (int nt = 0; nt < 16; ++nt) s[nt][r] *= inv;
    }

    // ---- O = P * V ----
    v8f oacc[4];
#pragma unroll
    for (int on = 0; on < 4; ++on) oacc[on] = zero8();

    for (int kc = 0; kc < 4; ++kc) {        // value rows kc*64 .. +63
        {
            // transpose-stage: vbufT[n][kk] = V[kc*64+kk][n]
            const float* g = v + (size_t)kc * 64 * 64 + (size_t)crow * 64 + ccol;
            float4 t0 = ((const float4*)g)[0], t1 = ((const float4*)g)[1];
            float4 t2 = ((const float4*)g)[2], t3 = ((const float4*)g)[3];
            vbufT[(ccol + 0)  * VST + crow] = t0.x;
            vbufT[(ccol + 1)  * VST + crow] = t0.y;
            vbufT[(ccol + 2)  * VST + crow] = t0.z;
            vbufT[(ccol + 3)  * VST + crow] = t0.w;
            vbufT[(ccol + 4)  * VST + crow] = t1.x;
            vbufT[(ccol + 5)  * VST + crow] = t1.y;
            vbufT[(ccol + 6)  * VST + crow] = t1.z;
            vbufT[(ccol + 7)  * VST + crow] = t1.w;
            vbufT[(ccol + 8)  * VST + crow] = t2.x;
            vbufT[(ccol + 9)  * VST + crow] = t2.y;
            vbufT[(ccol + 10) * VST + crow] = t2.z;
            vbufT[(ccol + 11) * VST + crow] = t2.w;
            vbufT[(ccol + 12) * VST + crow] = t3.x;
            vbufT[(ccol + 13) * VST + crow] = t3.y;
            vbufT[(ccol + 14) * VST + crow] = t3.z;
            vbufT[(ccol + 15) * VST + crow] = t3.w;
        }
        __syncthreads();
#pragma unroll
        for (int ntl = 0; ntl < 4; ++ntl) {
            const int nt = kc * 4 + ntl;
            // stage P tile (C/D layout -> A layout) through padded LDS
#pragma unroll
            for (int vv = 0; vv < 8; ++vv)
                pw[(kh * 8 + vv) * PST + l16] = s[nt][vv];
            __syncthreads();
#pragma unroll
            for (int ks = 0; ks < 4; ++ks) {
                v2f a = *(const v2f*)(pw + l16 * PST + ks * 4 + 2 * kh);
#pragma unroll
                for (int on = 0; on < 4; ++on) {
                    v2f bf = *(const v2f*)(vbufT + (on * 16 + l16) * VST +
                                           ntl * 16 + ks * 4 + 2 * kh);
                    oacc[on] = wmma_f32(a, bf, oacc[on]);
                }
            }
            __syncthreads();                // guards pbuf/vbufT reuse
        }
    }

    // store O as [b][t][j][512], dim index = h*64 + n
#pragma unroll
    for (int on = 0; on < 4; ++on) {
        const int n = on * 16 + l16;
#pragma unroll
        for (int vv = 0; vv < 8; ++vv) {
            const int t = m0 + kh * 8 + vv;
            O[(((size_t)b * 256 + t) * 24 + j) * 512 + h * 64 + n] = oacc[on][vv];
        }
    }
}

// ---------------------------------------------------------------------------
// Kernel 3: y[r][e] = sum_d O[r][d] * Wo[e][d] + bo[e]
// grid = (8, 384), block = 256. Wo rows are already K-major: pure row copy
// into stride-36 panel; fragments are contiguous b64.
// ---------------------------------------------------------------------------
__global__ void __launch_bounds__(256)
out_proj_kernel(const float* __restrict__ Oin, const float* __restrict__ Wo,
                const float* __restrict__ bo, float* __restrict__ y) {
    __shared__ float wbufT[64 * TST];   // 9.2 KB, [col n][kk]

    const int tid  = threadIdx.x;
    const int lane = tid & 31;
    const int wave = tid >> 5;
    const int kh   = lane >> 4;
    const int l16  = lane & 15;

    const int n0 = blockIdx.x * 64;
    const int m0 = blockIdx.y * 128 + wave * 16;

    const float* arow = Oin + (size_t)(m0 + l16) * 512;

    v8f acc[4];
#pragma unroll
    for (int on = 0; on < 4; ++on) acc[on] = zero8();

    const int ccol = tid >> 2;        // 0..63 : output column within panel
    const int ckq  = (tid & 3) * 8;   // 0..24 : kk group of 8

    for (int kc = 0; kc < 512; kc += 32) {
        // straight copy: wbufT[col][kk] = Wo[n0+col][kc+kk]
        {
            const float* g = Wo + (size_t)(n0 + ccol) * 512 + kc + ckq;
            float* d = wbufT + ccol * TST + ckq;
            *(float4*)(d)     = ((const float4*)g)[0];
            *(float4*)(d + 4) = ((const float4*)g)[1];
        }
        __syncthreads();
#pragma unroll
        for (int kr = 0; kr < 32; kr += 4) {
            v2f a = *(const v2f*)(arow + kc + kr + 2 * kh);
#pragma unroll
            for (int on = 0; on < 4; ++on) {
                v2f b = *(const v2f*)(wbufT + (on * 16 + l16) * TST + kr + 2 * kh);
                acc[on] = wmma_f32(a, b, acc[on]);
            }
        }
        __syncthreads();
    }

#pragma unroll
    for (int on = 0; on < 4; ++on) {
        const int col = n0 + on * 16 + l16;
        const float bias = bo[col];
#pragma unroll
        for (int vv = 0; vv < 8; ++vv)
            y[(size_t)(m0 + kh * 8 + vv) * 512 + col] = acc[on][vv] + bias;
    }
}

// ---------------------------------------------------------------------------
extern "C" void kernel_launch(void* const* d_in, const int* in_sizes, int n_in,
                              void* d_out, int out_size, void* d_ws, size_t ws_size,
                              hipStream_t stream) {
    const float* x  = (const float*)d_in[0];
    const float* Wq = (const float*)d_in[1];
    const float* Wk = (const float*)d_in[2];
    const float* Wv = (const float*)d_in[3];
    const float* Wo = (const float*)d_in[4];
    const float* bo = (const float*)d_in[5];
    float* out = (float*)d_out;

    const size_t TENSOR = (size_t)8 * 24 * 8 * 256 * 64;  // 25,165,824 floats
    float* ws = (float*)d_ws;
    float* Qw = ws;
    float* Kw = ws + TENSOR;
    float* Vw = ws + 2 * TENSOR;
    float* Ow = ws + 3 * TENSOR;

    dim3 blk(256);

    dim3 g1(512 / 64, 2048 / 128, 24);
    qkv_proj_kernel<<<g1, blk, 0, stream>>>(x, Wq, Qw);
    qkv_proj_kernel<<<g1, blk, 0, stream>>>(x, Wk, Kw);
    qkv_proj_kernel<<<g1, blk, 0, stream>>>(x, Wv, Vw);

    attn_kernel<<<dim3(2 * 8 * 24 * 8), blk, 0, stream>>>(Qw, Kw, Vw, Ow);

    out_proj_kernel<<<dim3(512 / 64, 49152 / 128), blk, 0, stream>>>(Ow, Wo, bo, out);
}